// MyModel_60292750901333
// MI455X (gfx1250) — compile-verified
//
#include <hip/hip_runtime.h>

typedef __attribute__((ext_vector_type(2))) float v2f;
typedef __attribute__((ext_vector_type(4))) float f4;
typedef __attribute__((ext_vector_type(8))) float v8f;

#define POLY_DIM 32

// out[b] = sum_{i=0}^{31} w[i] * x[b]^i
// WMMA f32 16x16x4: D[m][n] = sum_k w[4m+k] * x_n^k  (m = 0..7, rows 8..15 zero)
// then out[n] = Horner over m with t = x_n^4 (7 FMAs per element).
__global__ __launch_bounds__(256) void poly32_wmma_kernel(
    const float* __restrict__ x, const float* __restrict__ w,
    float* __restrict__ out, int n)
{
    const int lane  = threadIdx.x & 31;
    const int col   = lane & 15;   // column (batch elem) / A-matrix row index
    const int khalf = lane >> 4;   // 0 -> lanes 0-15 (K=0/1), 1 -> lanes 16-31 (K=2/3)

    // ---- A operand (16x4 f32, 2 VGPRs): VGPR0 = {K=0 | K=2}, VGPR1 = {K=1 | K=3}
    // A[m][k] = w[4m+k] for m=0..7, zero for m=8..15  (loaded once, divergence
    // reconverges before any WMMA).
    float a0 = 0.0f, a1 = 0.0f;
    if (col < 8) {
        a0 = w[4 * col + 2 * khalf + 0];
        a1 = w[4 * col + 2 * khalf + 1];
    }
    v2f A; A.x = a0; A.y = a1;

    const long wave  = (long)blockIdx.x * (blockDim.x >> 5) + (threadIdx.x >> 5);
    const long nwave = (long)gridDim.x * (blockDim.x >> 5);
    const long ntile = (long)n >> 6;              // 64 elements per wave-iteration

    for (long t = wave; t < ntile; t += nwave) {  // wave-uniform loop: EXEC all 1s
        const long base = t << 6;
        // Lanes n and n+16 read the same float4 (B column data needed in both
        // K-halves); unique traffic per wave = 256B in via B128 loads.
        const f4 xv = *(const f4*)(x + base + 4 * col);

        f4 res;
        #pragma unroll
        for (int j = 0; j < 4; ++j) {
            const float xj = xv[j];
            const float x2 = xj * xj;
            // ---- B operand (4x16 f32): VGPR0 = {K=0 | K=2}, VGPR1 = {K=1 | K=3}
            v2f B;
            B.x = khalf ? x2      : 1.0f;   // K=0 row: 1   | K=2 row: x^2
            B.y = khalf ? x2 * xj : xj;     // K=1 row: x   | K=3 row: x^3
            v8f C = {};
            // 8 args: (neg_a, A, neg_b, B, c_mod, C, reuse_a, reuse_b)
            C = __builtin_amdgcn_wmma_f32_16x16x4_f32(
                    false, A, false, B, (short)0, C, false, false);
            // D layout: VGPR r, lanes 0-15 hold M=r, N=lane -> lane `col` owns
            // D[0..7][col] in C[0..7]. Combine with t4 = x^4 Horner.
            const float t4 = x2 * x2;
            float r = C[7];
            #pragma unroll
            for (int m = 6; m >= 0; --m) r = r * t4 + C[m];
            res[j] = r;
        }
        if (khalf == 0) {                       // lanes 0-15 own the 64 results
            *(f4*)(out + base + 4 * col) = res; // B128 store, 256B per wave
        }
    }

    // Tail (n % 64): scalar Horner, no WMMA under divergent EXEC.
    const long tailBase = ntile << 6;
    if (wave == 0) {
        for (long i = tailBase + lane; i < (long)n; i += 32) {
            const float xi = x[i];
            float r = w[POLY_DIM - 1];
            for (int k = POLY_DIM - 2; k >= 0; --k) r = r * xi + w[k];
            out[i] = r;
        }
    }
}

extern "C" void kernel_launch(void* const* d_in, const int* in_sizes, int n_in,
                              void* d_out, int out_size, void* d_ws, size_t ws_size,
                              hipStream_t stream) {
    const float* x = (const float*)d_in[0];
    const float* w = (const float*)d_in[1];
    float* out = (float*)d_out;
    const int n = in_sizes[0];   // 4194304

    dim3 block(256);             // 8 wave32 waves per block
    dim3 grid(2048);             // 16384 waves -> 4 tiles of 64 elems per wave
    hipLaunchKernelGGL(poly32_wmma_kernel, grid, block, 0, stream, x, w, out, n);
}